// SqueezeExcitation_64218351010265
// MI455X (gfx1250) — compile-verified
//
#include <hip/hip_runtime.h>

// ---------------------------------------------------------------------------
// Squeeze-Excitation, MI455X (gfx1250, wave32).
//   x:[32,768,56,56] f32 -> pool -> MLP (WMMA f32 16x16x4) -> scale
// Memory-bound: ~924 MB HBM traffic => ~40us floor @ 23.3 TB/s.
// ---------------------------------------------------------------------------

typedef float v2f __attribute__((ext_vector_type(2)));
typedef float v8f __attribute__((ext_vector_type(8)));

#define C_DIM   768
#define CR_DIM  32
#define HW      3136      // 56*56
#define HW4     784       // HW/4 (float4 elements per plane)
#define NPLANE  24576     // 32 * 768

// ---------------------------------------------------------------------------
// Phase 1: global average pool. One block per (b,c) plane, float4 loads.
// ---------------------------------------------------------------------------
__global__ __launch_bounds__(256) void se_pool_kernel(
    const float* __restrict__ x, float* __restrict__ s) {
  const int plane = blockIdx.x;
  const float4* xp = (const float4*)(x + (size_t)plane * HW);
  float acc = 0.0f;
  for (int i = threadIdx.x; i < HW4; i += 256) {
    float4 v = xp[i];
    acc += (v.x + v.y) + (v.z + v.w);
  }
  __shared__ float red[8];
  #pragma unroll
  for (int off = 16; off > 0; off >>= 1) acc += __shfl_down(acc, off, 32);
  if ((threadIdx.x & 31) == 0) red[threadIdx.x >> 5] = acc;
  __syncthreads();
  if (threadIdx.x == 0) {
    float t = 0.0f;
    #pragma unroll
    for (int w = 0; w < 8; w++) t += red[w];
    s[plane] = t * (1.0f / (float)HW);
  }
}

// ---------------------------------------------------------------------------
// Phase 2: excitation MLP with V_WMMA_F32_16X16X4_F32. One block, 8 waves.
//   h = silu(s @ w1^T + b1)   [32 x 32]
//   g = sigmoid(h @ w2^T + b2) [32 x 768]
// GEMM1: K=768 split across 8 waves (96 each), partial C reduced via LDS.
// GEMM2: 48 N-tiles split 6 per wave; A (=h) kept in registers in A-layout.
// ---------------------------------------------------------------------------
__device__ __forceinline__ float fast_sigmoid(float v) {
  return 1.0f / (1.0f + __expf(-v));
}

__global__ __launch_bounds__(256) void se_mlp_kernel(
    const float* __restrict__ s,  const float* __restrict__ w1,
    const float* __restrict__ b1, const float* __restrict__ w2,
    const float* __restrict__ b2, float* __restrict__ g) {
  const int tid   = threadIdx.x;
  const int wave  = tid >> 5;          // 0..7
  const int lane  = tid & 31;          // wave32 lane
  const int ln15  = lane & 15;
  const int khalf = (lane >> 4) << 1;  // lanes 16-31 hold K+2,K+3 (A & B layout)
  const int mhi   = (lane >> 4) * 8;   // C/D layout: high lanes hold M=v+8

  __shared__ float partial[8][32][32]; // per-wave partial C of GEMM1 (32 KB)
  __shared__ float hlds[32][33];       // h matrix, padded stride (bank-conflict free)

  // ---- GEMM1: acc[mt][nt] += A(16x4 of s) * B(4x16 of w1^T), K slice per wave
  v8f acc[2][2] = {};
  const int k0 = wave * 96;
  for (int k = k0; k < k0 + 96; k += 4) {
    const int kb = k + khalf;
    v2f a0, a1, bb0, bb1;
    // A layout (16x4 f32): lane = M row, v2f = {K+0,K+1} (low) / {K+2,K+3} (high)
    a0.x  = s[(size_t)(ln15     ) * C_DIM + kb];
    a0.y  = s[(size_t)(ln15     ) * C_DIM + kb + 1];
    a1.x  = s[(size_t)(ln15 + 16) * C_DIM + kb];
    a1.y  = s[(size_t)(ln15 + 16) * C_DIM + kb + 1];
    // B layout (4x16 f32): VGPR0 = rows K+0 (low lanes)/K+2 (high), VGPR1 = K+1/K+3
    // B[k][n] = w1[n][k]
    bb0.x = w1[(size_t)(ln15     ) * C_DIM + kb];
    bb0.y = w1[(size_t)(ln15     ) * C_DIM + kb + 1];
    bb1.x = w1[(size_t)(ln15 + 16) * C_DIM + kb];
    bb1.y = w1[(size_t)(ln15 + 16) * C_DIM + kb + 1];
    acc[0][0] = __builtin_amdgcn_wmma_f32_16x16x4_f32(false, a0, false, bb0, (short)0, acc[0][0], false, false);
    acc[0][1] = __builtin_amdgcn_wmma_f32_16x16x4_f32(false, a0, false, bb1, (short)0, acc[0][1], false, false);
    acc[1][0] = __builtin_amdgcn_wmma_f32_16x16x4_f32(false, a1, false, bb0, (short)0, acc[1][0], false, false);
    acc[1][1] = __builtin_amdgcn_wmma_f32_16x16x4_f32(false, a1, false, bb1, (short)0, acc[1][1], false, false);
  }
  // Scatter partial C to LDS per C/D layout: VGPR v -> M = mt*16 + v + mhi
  #pragma unroll
  for (int mt = 0; mt < 2; mt++)
    #pragma unroll
    for (int nt = 0; nt < 2; nt++)
      #pragma unroll
      for (int v = 0; v < 8; v++)
        partial[wave][mt * 16 + v + mhi][nt * 16 + ln15] = acc[mt][nt][v];
  __syncthreads();

  // ---- Reduce 8 partials, add bias, SiLU -> h in LDS
  for (int i = tid; i < 32 * 32; i += 256) {
    const int r = i >> 5, c = i & 31;
    float sum = 0.0f;
    #pragma unroll
    for (int w = 0; w < 8; w++) sum += partial[w][r][c];
    sum += b1[c];
    hlds[r][c] = sum * fast_sigmoid(sum);   // SiLU
  }
  __syncthreads();

  // ---- GEMM2: g = sigmoid(h @ w2^T + b2), M=32, K=32, N=768
  // Preload h into A-register layout (2 M-tiles x 8 K-steps, 32 floats/lane)
  v2f ha[2][8];
  #pragma unroll
  for (int mt = 0; mt < 2; mt++)
    #pragma unroll
    for (int ks = 0; ks < 8; ks++) {
      ha[mt][ks].x = hlds[mt * 16 + ln15][ks * 4 + khalf];
      ha[mt][ks].y = hlds[mt * 16 + ln15][ks * 4 + khalf + 1];
    }

  const int nt0 = wave * 6;                 // 48 N-tiles / 8 waves
  for (int nt = nt0; nt < nt0 + 6; nt++) {
    const int N = nt * 16 + ln15;           // output channel 0..767
    v8f c0 = {}, c1 = {};
    #pragma unroll
    for (int ks = 0; ks < 8; ks++) {
      const int kb = ks * 4 + khalf;
      v2f bb;                               // B[k][n] = w2[n][k], w2 is [768,32]
      bb.x = w2[(size_t)N * CR_DIM + kb];
      bb.y = w2[(size_t)N * CR_DIM + kb + 1];
      c0 = __builtin_amdgcn_wmma_f32_16x16x4_f32(false, ha[0][ks], false, bb, (short)0, c0, false, false);
      c1 = __builtin_amdgcn_wmma_f32_16x16x4_f32(false, ha[1][ks], false, bb, (short)0, c1, false, false);
    }
    const float bias = b2[N];
    #pragma unroll
    for (int v = 0; v < 8; v++) {
      g[(size_t)(     v + mhi) * C_DIM + N] = fast_sigmoid(c0[v] + bias);
      g[(size_t)(16 + v + mhi) * C_DIM + N] = fast_sigmoid(c1[v] + bias);
    }
  }
}

// ---------------------------------------------------------------------------
// Phase 3: out = x * g (per-plane scalar gate). One block per plane, float4.
// ---------------------------------------------------------------------------
__global__ __launch_bounds__(256) void se_scale_kernel(
    const float* __restrict__ x, const float* __restrict__ g,
    float* __restrict__ out) {
  const int plane = blockIdx.x;
  const float gate = g[plane];
  const float4* xp = (const float4*)(x + (size_t)plane * HW);
  float4* op = (float4*)(out + (size_t)plane * HW);
  for (int i = threadIdx.x; i < HW4; i += 256) {
    float4 v = xp[i];
    v.x *= gate; v.y *= gate; v.z *= gate; v.w *= gate;
    op[i] = v;
  }
}

// ---------------------------------------------------------------------------
extern "C" void kernel_launch(void* const* d_in, const int* in_sizes, int n_in,
                              void* d_out, int out_size, void* d_ws, size_t ws_size,
                              hipStream_t stream) {
  const float* x  = (const float*)d_in[0];   // [32,768,56,56]
  const float* w1 = (const float*)d_in[1];   // [32,768]
  const float* b1 = (const float*)d_in[2];   // [32]
  const float* w2 = (const float*)d_in[3];   // [768,32]
  const float* b2 = (const float*)d_in[4];   // [768]
  float* out = (float*)d_out;

  float* s = (float*)d_ws;                   // [32,768] pooled means
  float* g = s + NPLANE;                     // [32,768] gates

  se_pool_kernel <<<NPLANE, 256, 0, stream>>>(x, s);
  se_mlp_kernel  <<<1,      256, 0, stream>>>(s, w1, b1, w2, b2, g);
  se_scale_kernel<<<NPLANE, 256, 0, stream>>>(x, g, out);
}